// GCNEncoder_60550448939586
// MI455X (gfx1250) — compile-verified
//
#include <hip/hip_runtime.h>

#define IN_C  256
#define HID   256
#define OUT_C 128

typedef __attribute__((ext_vector_type(2))) float v2f;
typedef __attribute__((ext_vector_type(8))) float v8f;

// ---------------------------------------------------------------------------
// Degree / normalization
// ---------------------------------------------------------------------------
__global__ void k_init_deg(float* __restrict__ deg, int N) {
  int i = blockIdx.x * blockDim.x + threadIdx.x;
  if (i < N) deg[i] = 1.0f;                       // self-loop contributes 1
}

__global__ void k_count_deg(const long long* __restrict__ dst,
                            float* __restrict__ deg, long long E) {
  long long e = (long long)blockIdx.x * blockDim.x + threadIdx.x;
  if (e < E) unsafeAtomicAdd(&deg[dst[e]], 1.0f); // exact: counts << 2^24
}

__global__ void k_rsqrt_deg(float* __restrict__ deg, int N) {
  int i = blockIdx.x * blockDim.x + threadIdx.x;
  if (i < N) {
    float d = deg[i];
    deg[i] = (d > 0.0f) ? rsqrtf(d) : 0.0f;       // deg -> dis, in place
  }
}

// ---------------------------------------------------------------------------
// FP32 WMMA GEMM: C[M x Nn] = A[M x K] * B[K x Nn], row-major.
// K/Nn are compile-time so B addressing strength-reduces to pointer increments
// with immediate offsets. One wave computes a 16 x NT tile; all NJ B-fragments
// of a K-step are fetched before the NJ V_WMMA_F32_16X16X4_F32 issues so the
// loadcnt waits stagger instead of serializing each WMMA on its own loads.
// A-frag (16x4):  lane m = lane%16, VGPR0 = A[m][k0+2*half], VGPR1 = +1
// B-frag (4x16):  lane n = lane%16, VGPR0 = B[k0+2*half][n], VGPR1 = +1
// C/D (16x16): VGPR v -> row v + 8*half, col = lane%16
// ---------------------------------------------------------------------------
template <int K, int Nn, int NT>   // NT = output columns per wave
__global__ void k_wmma_gemm_f32(const float* __restrict__ A,
                                const float* __restrict__ B,
                                float* __restrict__ C, int M) {
  const int wavesPerBlock = blockDim.x >> 5;
  const int wid  = threadIdx.x >> 5;
  const int lane = threadIdx.x & 31;
  const int half = lane >> 4;
  const int lcol = lane & 15;
  constexpr int tilesN = Nn / NT;
  const int numTiles = (M >> 4) * tilesN;

  int tile = blockIdx.x * wavesPerBlock + wid;     // wave-uniform
  if (tile >= numTiles) return;                    // uniform per wave: EXEC stays full
  const int tm = (tile / tilesN) << 4;
  const int tn = (tile % tilesN) * NT;

  constexpr int NJ = NT / 16;
  v8f acc[NJ];
  #pragma unroll
  for (int j = 0; j < NJ; ++j)
    acc[j] = (v8f){0.f, 0.f, 0.f, 0.f, 0.f, 0.f, 0.f, 0.f};

  const float* __restrict__ Aptr = A + (size_t)(tm + lcol) * K + 2 * half;
  const float* __restrict__ Bptr = B + (size_t)(2 * half) * Nn + tn + lcol;

  for (int k0 = 0; k0 < K; k0 += 4) {
    const v2f a = *(const v2f*)Aptr;               // 8B aligned
    v2f b[NJ];
    #pragma unroll
    for (int j = 0; j < NJ; ++j) {                 // fetch all B frags first
      b[j].x = Bptr[j * 16];                       // row k0+2*half
      b[j].y = Bptr[Nn + j * 16];                  // row k0+2*half+1
    }
    #pragma unroll
    for (int j = 0; j < NJ; ++j) {
      acc[j] = __builtin_amdgcn_wmma_f32_16x16x4_f32(
          /*neg_a=*/false, a, /*neg_b=*/false, b[j],
          /*c_mod=*/(short)0, acc[j], /*reuse_a=*/false, /*reuse_b=*/false);
    }
    Aptr += 4;
    Bptr += (size_t)4 * Nn;
  }

  #pragma unroll
  for (int j = 0; j < NJ; ++j) {
    float* __restrict__ Crow =
        C + (size_t)(tm + half * 8) * Nn + tn + j * 16 + lcol;
    #pragma unroll
    for (int v = 0; v < 8; ++v) Crow[(size_t)v * Nn] = acc[j][v];
  }
}

// ---------------------------------------------------------------------------
// out[i][c] = (useBias ? bias[c] : 0) + dis[i]^2 * h[i][c]   (self-loop term)
// ---------------------------------------------------------------------------
__global__ void k_self_init(const float* __restrict__ h,
                            const float* __restrict__ dis,
                            const float* __restrict__ bias,
                            float* __restrict__ out,
                            long long total, int C, int useBias) {
  long long idx = (long long)blockIdx.x * blockDim.x + threadIdx.x;
  if (idx >= total) return;
  int c = (int)(idx % C);
  long long i = idx / C;
  float d = dis[i];
  float v = d * d * h[idx];
  if (useBias) v += bias[c];
  out[idx] = v;
}

// ---------------------------------------------------------------------------
// agg[i][c] = relu(agg[i][c] + bias[c])
// ---------------------------------------------------------------------------
__global__ void k_bias_relu(float* __restrict__ a, const float* __restrict__ bias,
                            long long total, int C) {
  long long idx = (long long)blockIdx.x * blockDim.x + threadIdx.x;
  if (idx >= total) return;
  float v = a[idx] + bias[(int)(idx % C)];
  a[idx] = v > 0.0f ? v : 0.0f;
}

// ---------------------------------------------------------------------------
// Edge scatter: one wave per edge. w = dis[src]*dis[dst];
// out[dst][c] += w * H[src][c]  via hardware global_atomic_add_f32.
// Gathers hit L2 (H is 51-102 MB < 192 MB global L2 on MI455X).
// ---------------------------------------------------------------------------
template <int CH>
__global__ void k_edge_scatter(const long long* __restrict__ src,
                               const long long* __restrict__ dst,
                               const float* __restrict__ dis,
                               const float* __restrict__ H,
                               float* __restrict__ out, long long E) {
  const int lane = threadIdx.x & 31;
  long long e = (long long)blockIdx.x * (blockDim.x >> 5) + (threadIdx.x >> 5);
  if (e >= E) return;

  const long long r = src[e];
  const long long c = dst[e];
  const float w = dis[r] * dis[c];

  const float* __restrict__ hrow = H + (size_t)r * CH;
  float* __restrict__ orow = out + (size_t)c * CH;

  constexpr int V4 = CH / 128;                    // float4 chunks per lane (2 or 1)
  #pragma unroll
  for (int i = 0; i < V4; ++i) {
    const int q = lane * V4 + i;                  // each lane reads V4 adjacent float4 chunks
    const float4 v = ((const float4*)hrow)[q];
    const int base = q * 4;
    unsafeAtomicAdd(orow + base + 0, w * v.x);
    unsafeAtomicAdd(orow + base + 1, w * v.y);
    unsafeAtomicAdd(orow + base + 2, w * v.z);
    unsafeAtomicAdd(orow + base + 3, w * v.w);
  }
}

// ---------------------------------------------------------------------------
// Orchestration
// ---------------------------------------------------------------------------
extern "C" void kernel_launch(void* const* d_in, const int* in_sizes, int n_in,
                              void* d_out, int out_size, void* d_ws, size_t ws_size,
                              hipStream_t stream) {
  const float*     x   = (const float*)d_in[0];
  const long long* ei  = (const long long*)d_in[1];   // int64 [2, E]
  const float*     W1  = (const float*)d_in[2];
  const float*     b1  = (const float*)d_in[3];
  const float*     W2  = (const float*)d_in[4];
  const float*     b2  = (const float*)d_in[5];
  float*           out = (float*)d_out;

  const int       N = in_sizes[0] / IN_C;             // 100000 (multiple of 16)
  const long long E = (long long)in_sizes[1] / 2;     // 3200000
  const long long* src = ei;        // edge_index[0] = source j
  const long long* dst = ei + E;    // edge_index[1] = target i

  // Workspace: dis[N] | h[N*HID] (reused for h2) | agg[N*HID]
  float* dis = (float*)d_ws;
  float* h   = dis + (((size_t)N + 255) & ~(size_t)255);
  float* agg = h + (size_t)N * HID;

  const int TB = 256;  // 8 wave32 waves per block

  // ---- normalization: deg -> dis = rsqrt(deg) ----
  k_init_deg<<<(N + TB - 1) / TB, TB, 0, stream>>>(dis, N);
  k_count_deg<<<(int)((E + TB - 1) / TB), TB, 0, stream>>>(dst, dis, E);
  k_rsqrt_deg<<<(N + TB - 1) / TB, TB, 0, stream>>>(dis, N);

  // ---- layer 1: h = x @ W1 ----
  {
    int tiles = (N >> 4) * (HID / 64);               // 25000 (16x64 tiles)
    k_wmma_gemm_f32<IN_C, HID, 64>
        <<<(tiles + 7) / 8, TB, 0, stream>>>(x, W1, h, N);
  }
  const long long tot1 = (long long)N * HID;
  k_self_init<<<(int)((tot1 + TB - 1) / TB), TB, 0, stream>>>(h, dis, b1, agg, tot1, HID, 0);
  k_edge_scatter<HID><<<(int)((E + 7) / 8), TB, 0, stream>>>(src, dst, dis, h, agg, E);
  k_bias_relu<<<(int)((tot1 + TB - 1) / TB), TB, 0, stream>>>(agg, b1, tot1, HID);

  // ---- layer 2: h2 = relu_agg @ W2 (h2 reuses h buffer) ----
  {
    int tiles = (N >> 4) * (OUT_C / 64);             // 12500 (16x64 tiles)
    k_wmma_gemm_f32<HID, OUT_C, 64>
        <<<(tiles + 7) / 8, TB, 0, stream>>>(agg, W2, h, N);
  }
  const long long tot2 = (long long)N * OUT_C;
  k_self_init<<<(int)((tot2 + TB - 1) / TB), TB, 0, stream>>>(h, dis, b2, out, tot2, OUT_C, 1);
  k_edge_scatter<OUT_C><<<(int)((E + 7) / 8), TB, 0, stream>>>(src, dst, dis, h, out, E);
}